// compEigDecomp_batch_14499809591881
// MI455X (gfx1250) — compile-verified
//
#include <hip/hip_runtime.h>
#include <hip/hip_bf16.h>

typedef float v2f __attribute__((ext_vector_type(2)));
typedef float v8f __attribute__((ext_vector_type(8)));
typedef unsigned int u32x4 __attribute__((ext_vector_type(4)));
typedef int i32x4 __attribute__((ext_vector_type(4)));
typedef int i32x8 __attribute__((ext_vector_type(8)));

#define TM 128
#define TN 64
#define KC 16
#define LP (KC + 4)  // LDS row stride 20 floats = 80B: 16B-aligned, no bank conflicts

__device__ __forceinline__ v8f wmma_f32_4(v2f a, v2f b, v8f c) {
  // V_WMMA_F32_16X16X4_F32 : D = A(16x4) * B(4x16) + C(16x16), fp32
  return __builtin_amdgcn_wmma_f32_16x16x4_f32(false, a, false, b, (short)0, c,
                                               false, false);
}

// Low 32 bits of a generic pointer to __shared__ = LDS byte offset (aperture
// check only looks at the high half; flat->LDS truncates to addr[31:0]).
__device__ __forceinline__ unsigned lds_addr(const void* p) {
  return (unsigned)(uintptr_t)p;
}

// Async global->LDS copy (ASYNCcnt-tracked, no VGPR staging)
__device__ __forceinline__ void async_copy_b32(unsigned lds, const void* g) {
  asm volatile("global_load_async_to_lds_b32 %0, %1, off" ::"v"(lds), "v"(g)
               : "memory");
}

// Tensor Data Mover: DMA a 2D tile (tile_x elems/row, tile_y rows) of a
// strided f32 tensor into LDS, with the D# pad feature inserting
// pad_amount(+1) DWORDs after every 2<<pad_interval DWORDs stored — used to
// reproduce the padded LDS row stride LP. Wave-level op, TENSORcnt-tracked.
__device__ __forceinline__ void tdm_load_2d_f32(unsigned lds_byte,
                                                const void* gaddr,
                                                unsigned tile_x,
                                                unsigned tile_y,
                                                unsigned tensor_w,
                                                unsigned tensor_h,
                                                unsigned row_stride_elems,
                                                unsigned pad_interval_code,
                                                unsigned pad_amount_code,
                                                bool pad_en) {
  unsigned long long ga = (unsigned long long)(uintptr_t)gaddr;
  u32x4 g0;
  g0.x = 1u;                                   // count=1, user descriptor
  g0.y = lds_byte;                             // lds_addr [63:32]
  g0.z = (unsigned)(ga & 0xffffffffu);         // global_addr lo
  g0.w = (unsigned)((ga >> 32) & 0x01ffffffu)  // global_addr [56:32]
         | (2u << 30);                         // type=2 ("image")
  i32x8 g1;
  g1[0] = (int)((2u << 16)                       // data_size=2 -> 4 bytes
                | (pad_en ? (1u << 20) : 0u)     // pad_enable
                | (pad_interval_code << 22)      // pad_interval
                | (pad_amount_code << 25));      // pad_amount
  g1[1] = (int)((tensor_w & 0xffffu) << 16);     // tensor_dim0[15:0] @ [63:48]
  g1[2] = (int)(((tensor_w >> 16) & 0xffffu) | ((tensor_h & 0xffffu) << 16));
  g1[3] = (int)(((tensor_h >> 16) & 0xffffu) | ((tile_x & 0xffffu) << 16));
  g1[4] = (int)(tile_y & 0xffffu);               // tile_dim1; tile_dim2=0
  g1[5] = (int)row_stride_elems;                 // tensor_dim0_stride[31:0]
  g1[6] = 0;                                     // stride hi / dim1_stride lo
  g1[7] = 0;
#if __has_builtin(__builtin_amdgcn_tensor_load_to_lds_d2)
  __builtin_amdgcn_tensor_load_to_lds_d2(g0, g1, 0);
#else
  i32x4 g2 = {1, 0, 0, 0};  // tensor_dim2=1; tile_dim3=0 (unused)
  i32x4 g3 = {0, 0, 0, 0};
  i32x8 g4 = {0, 0, 0, 0, 0, 0, 0, 0};
  __builtin_amdgcn_tensor_load_to_lds(g0, g1, g2, g3, g4, 0);
#endif
}

// C[b] = A[b] (M x K, lda) * B[b] (K x N, ldb), block tile 128x64.
// A tiles staged by the TDM (wave 0 issues one descriptor per tile),
// B tiles transposed into LDS via per-lane async b32 copies; both double-
// buffered. grid = (M/128, N/64, batch)
__global__ __launch_bounds__(256) void gemm_wmma_f32_kernel(
    const float* __restrict__ A, const float* __restrict__ B,
    float* __restrict__ C, int K, int lda, int ldb, int ldc,
    long strideA, long strideB, long strideC) {
  __shared__ float As[2][TM][LP];   // A tile, row-major (TDM pad -> stride LP)
  __shared__ float BsT[2][TN][LP];  // B tile, TRANSPOSED: [col][k]

  const int tid = threadIdx.x;
  const int m0 = blockIdx.x * TM;
  const int n0 = blockIdx.y * TN;

  const float* Ab = A + (long)blockIdx.z * strideA;
  const float* Bb = B + (long)blockIdx.z * strideB;
  float* Cb = C + (long)blockIdx.z * strideC;

  const int wave = tid >> 5;
  const int lane = tid & 31;
  const int hl = lane >> 4;  // lane-half: K pair (A/B frags) / row+8 (C frag)
  const int lr = lane & 15;
  const int wm = (wave & 3) * 32;   // 4 waves along M
  const int wn = (wave >> 2) * 32;  // 2 waves along N

  v8f acc[2][2];
#pragma unroll
  for (int mi = 0; mi < 2; ++mi)
#pragma unroll
    for (int ni = 0; ni < 2; ++ni) acc[mi][ni] = (v8f){0, 0, 0, 0, 0, 0, 0, 0};

  // One tile's staging: wave0 issues 1 TDM op for A (TENSORcnt), every wave
  // issues 4 async b32 copies scattering B into transposed LDS (ASYNCcnt).
  auto issue_tile = [&](int kt, int buf) {
    const int k = kt * KC;
    if (wave == 0) {
      // pad_interval=3 -> pad after every 16 DWORDs (one 16-float tile row);
      // pad_amount=3 -> skip 4 DWORDs: reproduces LDS row stride LP=20.
      tdm_load_2d_f32(lds_addr(&As[buf][0][0]), Ab + (long)m0 * lda + k,
                      /*tile_x=*/KC, /*tile_y=*/TM,
                      /*tensor_w=*/(unsigned)lda, /*tensor_h=*/0x100000u,
                      /*row_stride=*/(unsigned)lda,
                      /*pad_interval=*/3u, /*pad_amount=*/3u, /*pad_en=*/true);
    }
#pragma unroll
    for (int i = 0; i < 4; ++i) {
      int e = tid + i * 256;  // 1024 floats = 16x64 B tile
      int grow = e >> 6;      // k within tile
      int gcol = e & 63;      // column
      async_copy_b32(lds_addr(&BsT[buf][gcol][grow]),
                     Bb + (long)(k + grow) * ldb + n0 + gcol);
    }
  };

  const int KT = K / KC;
  issue_tile(0, 0);

  for (int kt = 0; kt < KT; ++kt) {
    const int buf = kt & 1;
    const bool pre = (kt + 1 < KT);
    if (pre) {
      issue_tile(kt + 1, buf ^ 1);  // prefetch next tile into other buffer
      if (wave == 0) __builtin_amdgcn_s_wait_tensorcnt(1);  // cur TDM done
      asm volatile("s_wait_asynccnt 0x4" ::: "memory");     // cur B done
    } else {
      if (wave == 0) __builtin_amdgcn_s_wait_tensorcnt(0);
      asm volatile("s_wait_asynccnt 0x0" ::: "memory");
    }
    __syncthreads();  // all waves' staged data visible

#pragma unroll
    for (int kk = 0; kk < KC; kk += 4) {
      // A frags (16x4): lanes 0-15 carry K=kk,kk+1; lanes 16-31 K=kk+2,kk+3
      // contiguous in LDS -> ds_load_b64
      v2f a0 = *(const v2f*)&As[buf][wm + lr][kk + 2 * hl];
      v2f a1 = *(const v2f*)&As[buf][wm + 16 + lr][kk + 2 * hl];
      // B frags (4x16): transposed LDS makes (K,K+1) contiguous -> ds_load_b64
      v2f b0 = *(const v2f*)&BsT[buf][wn + lr][kk + 2 * hl];
      v2f b1 = *(const v2f*)&BsT[buf][wn + 16 + lr][kk + 2 * hl];

      acc[0][0] = wmma_f32_4(a0, b0, acc[0][0]);
      acc[0][1] = wmma_f32_4(a0, b1, acc[0][1]);
      acc[1][0] = wmma_f32_4(a1, b0, acc[1][0]);
      acc[1][1] = wmma_f32_4(a1, b1, acc[1][1]);
    }
    __syncthreads();  // everyone done reading buf before it is overwritten
  }

  // C 16x16 layout: VGPR i -> row (i + 8*half), col = lane%16
#pragma unroll
  for (int mi = 0; mi < 2; ++mi)
#pragma unroll
    for (int ni = 0; ni < 2; ++ni)
#pragma unroll
      for (int i = 0; i < 8; ++i) {
        int row = m0 + wm + mi * 16 + i + 8 * hl;
        int col = n0 + wn + ni * 16 + lr;
        Cb[(long)row * ldc + col] = acc[mi][ni][i];
      }
}

__global__ void fill_zero_kernel(float* p, long n) {
  long i = (long)blockIdx.x * blockDim.x + threadIdx.x;
  long stride = (long)gridDim.x * blockDim.x;
  for (; i < n; i += stride) p[i] = 0.0f;
}

// Reproduce the one-hot probe matrix from the reference (ns=49 hardcoded)
__global__ void set_probes_kernel(float* X, const int* rp, const int* dp,
                                  int w, int ldx, long stride, int bs) {
  const int ns0 = 49;
  int r = rp[0], d = dp[0];
  float wint = sqrtf((float)w / (float)ns0);
  int nW = (int)roundf((float)r / wint);
  int nH = (int)roundf((float)ns0 / (float)nW);
  float wintf = (float)r / (float)nW;
  float hintf = (float)d / (float)nH;
  int ns = nW * nH;
  int c = threadIdx.x;
  if (c < ns) {
    int j = c / nH;
    int i = c % nH;
    int pos =
        (int)roundf((j + 0.5f) * wintf) * d + (int)roundf((i + 0.5f) * hintf);
    for (int b = 0; b < bs; ++b)
      X[(long)b * stride + (long)pos * ldx + c] = 1.0f;
  }
}

// Modified Gram-Schmidt QR of thin [w, ns] (leading dim ldx); Q overwrites X,
// |R_jj| -> t0. One workgroup per batch.
__global__ __launch_bounds__(256) void qr_mgs_kernel(float* X, float* t0,
                                                     int w, int ns, int ldx,
                                                     long stride) {
  __shared__ float red[256];
  float* Xb = X + (long)blockIdx.x * stride;
  const int tid = threadIdx.x;
  for (int j = 0; j < ns; ++j) {
    float s = 0.0f;
    for (int r = tid; r < w; r += 256) {
      float v = Xb[(long)r * ldx + j];
      s += v * v;
    }
    red[tid] = s;
    __syncthreads();
    for (int off = 128; off > 0; off >>= 1) {
      if (tid < off) red[tid] += red[tid + off];
      __syncthreads();
    }
    float nrm = sqrtf(red[0]);
    __syncthreads();
    if (tid == 0) t0[blockIdx.x * ns + j] = fabsf(nrm);
    float inv = (nrm > 0.0f) ? (1.0f / nrm) : 0.0f;
    for (int r = tid; r < w; r += 256) Xb[(long)r * ldx + j] *= inv;
    __syncthreads();
    for (int k = j + 1; k < ns; ++k) {
      float dsum = 0.0f;
      for (int r = tid; r < w; r += 256)
        dsum += Xb[(long)r * ldx + j] * Xb[(long)r * ldx + k];
      red[tid] = dsum;
      __syncthreads();
      for (int off = 128; off > 0; off >>= 1) {
        if (tid < off) red[tid] += red[tid + off];
        __syncthreads();
      }
      float rjk = red[0];
      __syncthreads();
      for (int r = tid; r < w; r += 256)
        Xb[(long)r * ldx + k] -= rjk * Xb[(long)r * ldx + j];
      __syncthreads();
    }
  }
}

// d_out = [t0 (bs*ns)] ++ [x (bs, w, ns)] stripping the 64-col padding
__global__ void writeout_kernel(const float* X, const float* t0, float* out,
                                int bs, int w, int ns, int ldx, long stride,
                                long total) {
  long i = (long)blockIdx.x * blockDim.x + threadIdx.x;
  long gstride = (long)gridDim.x * blockDim.x;
  long nhead = (long)bs * ns;
  for (; i < total; i += gstride) {
    if (i < nhead) {
      out[i] = t0[i];
    } else {
      long q = i - nhead;
      long perb = (long)w * ns;
      int b = (int)(q / perb);
      long rem = q % perb;
      int row = (int)(rem / ns);
      int c = (int)(rem % ns);
      out[i] = X[(long)b * stride + (long)row * ldx + c];
    }
  }
}

extern "C" void kernel_launch(void* const* d_in, const int* in_sizes, int n_in,
                              void* d_out, int out_size, void* d_ws,
                              size_t ws_size, hipStream_t stream) {
  const float* In = (const float*)d_in[0];
  const int* rp = (const int*)d_in[1];
  const int* dp = (const int*)d_in[2];
  float* out = (float*)d_out;

  const int bs = 8, w = 4096, ns = 49, ldx = 64;
  const long wxw = (long)w * w;
  const long xstride = (long)w * ldx;

  char* ws = (char*)d_ws;
  float* A = (float*)ws;                            // 8*4096*4096 f32 = 512MB
  float* xa = (float*)(ws + (size_t)bs * wxw * 4);  // 8*4096*64 f32
  float* xb = xa + (size_t)bs * xstride;
  float* t0 = xb + (size_t)bs * xstride;            // 8*49 f32

  // A = In @ In (batched, fp32 WMMA)
  dim3 gridA(w / TM, w / TN, bs);
  gemm_wmma_f32_kernel<<<gridA, 256, 0, stream>>>(In, In, A, w, w, w, w, wxw,
                                                  wxw, wxw);

  // x0 = probe matrix (padded to 64 cols)
  fill_zero_kernel<<<2048, 256, 0, stream>>>(xa, (long)bs * xstride);
  set_probes_kernel<<<1, 64, 0, stream>>>(xa, rp, dp, w, ldx, xstride, bs);

  // 2 outer iters: 20 power steps (full-N 128x64 blocks: A streamed exactly
  // once per step) then MGS QR in place
  dim3 gridP(w / TM, 1, bs);
  for (int outer = 0; outer < 2; ++outer) {
    float* cur = xa;
    float* nxt = xb;
    for (int it = 0; it < 20; ++it) {
      gemm_wmma_f32_kernel<<<gridP, 256, 0, stream>>>(
          A, cur, nxt, w, w, ldx, ldx, wxw, xstride, xstride);
      float* t = cur;
      cur = nxt;
      nxt = t;
    }
    // 20 steps (even) -> result back in xa
    qr_mgs_kernel<<<bs, 256, 0, stream>>>(xa, t0, w, ns, ldx, xstride);
  }

  long total = (long)bs * ns + (long)bs * w * ns;
  writeout_kernel<<<4096, 256, 0, stream>>>(xa, t0, out, bs, w, ns, ldx,
                                            xstride, total);
}